// KDChannelLoss_57767310131635
// MI455X (gfx1250) — compile-verified
//
#include <hip/hip_runtime.h>
#include <hip/hip_bf16.h>

// KD channel loss: S[i,c] = sum_p gt[i,p]*src[c,p] (and tgt), counts[i] = sum_p gt[i,p]
// then tiny symmetric-KL on 4x4 row-softmaxes. Memory-bound: 192 MiB streamed once.
//
// GEMM mapping onto V_WMMA_F32_16X16X4_F32 (wave32):
//   16x16 tile = 4x4 grid of 4x4 blocks; row M=4g+i, col N=4s+c, K=kk (4 pixels).
//   Pixel of (g,kk) = tilebase + 4g + kk. Diagonal blocks (g==s) hold the real
//   sums over 16 pixels; off-diagonal blocks are never read. One WMMA eats 16 px.

typedef __attribute__((ext_vector_type(2))) float v2f;
typedef __attribute__((ext_vector_type(4))) float f4;
typedef __attribute__((ext_vector_type(8))) float v8f;

#define KD_HW    (512 * 512)       // 2^18
#define KD_NPIX  (16 * KD_HW)      // 2^22  (B*H*W)
#define KD_BLOCKS  512
#define KD_THREADS 256

// swap wave halves: ds_swizzle group-of-32, xor=0x10, and=0x1f -> offset 0x401F
__device__ __forceinline__ float kd_swz16(float x) {
    int r = __builtin_amdgcn_ds_swizzle(__builtin_bit_cast(int, x), 0x401F);
    return __builtin_bit_cast(float, r);
}

__global__ void kd_zero_ws(float* ws) {
    if (threadIdx.x < 36) ws[threadIdx.x] = 0.0f;
}

__global__ __launch_bounds__(KD_THREADS)
void kd_sums_kernel(const float* __restrict__ src, const float* __restrict__ tgt,
                    const float* __restrict__ gt, float* __restrict__ ws) {
    __shared__ float sred[36];
    const int tid = threadIdx.x;
    if (tid < 36) sred[tid] = 0.0f;
    __syncthreads();

    const int  lane = tid & 31;
    const bool hi   = (lane & 16) != 0;
    const int  ch   = lane & 3;           // channel this lane streams
    const int  grp  = (lane >> 2) & 7;    // which float4 within the 32-pixel span

    const unsigned gwave = blockIdx.x * (KD_THREADS / 32) + (tid >> 5);
    const unsigned nwave = KD_BLOCKS * (KD_THREADS / 32);
    const unsigned pstep = nwave * 32u;

    v8f accS = {};   // source accumulator (16x16 f32, diag blocks = real sums)
    v8f accT = {};   // target accumulator
    float cnt = 0.0f;

    for (unsigned p0 = gwave * 32u; p0 < KD_NPIX; p0 += pstep) {
        const unsigned b  = p0 >> 18;            // batch index
        const unsigned hw = p0 & (KD_HW - 1u);   // pixel within image (32-aligned)
        const size_t base = (size_t)b * (4u * KD_HW) + (size_t)ch * KD_HW
                          + (size_t)hw + 4u * (unsigned)grp;

        // coalesced, streamed-once (non-temporal) 128-bit loads
        const f4 qg = __builtin_nontemporal_load((const f4*)(gt  + base));
        const f4 qs = __builtin_nontemporal_load((const f4*)(src + base));
        const f4 qt = __builtin_nontemporal_load((const f4*)(tgt + base));

        cnt += qg.x + qg.y + qg.z + qg.w;        // per-class pixel counts

        // Fragment fixup: lanes 0-15 hold tile0 pixels, lanes 16-31 hold tile1.
        // A/B frag for a tile needs hi lanes to carry K=2,3 of the *same* tile,
        // so swap halves and select.
        const float gx = kd_swz16(qg.x), gy = kd_swz16(qg.y),
                    gz = kd_swz16(qg.z), gw = kd_swz16(qg.w);
        const float sx = kd_swz16(qs.x), sy = kd_swz16(qs.y),
                    sz = kd_swz16(qs.z), sw = kd_swz16(qs.w);
        const float tx = kd_swz16(qt.x), ty = kd_swz16(qt.y),
                    tz = kd_swz16(qt.z), tw = kd_swz16(qt.w);

        v2f A0, A1, Bs0, Bs1, Bt0, Bt1;
        A0.x  = hi ? gz   : qg.x;  A0.y  = hi ? gw   : qg.y;   // gt, tile0
        A1.x  = hi ? qg.z : gx;    A1.y  = hi ? qg.w : gy;     // gt, tile1
        Bs0.x = hi ? sz   : qs.x;  Bs0.y = hi ? sw   : qs.y;
        Bs1.x = hi ? qs.z : sx;    Bs1.y = hi ? qs.w : sy;
        Bt0.x = hi ? tz   : qt.x;  Bt0.y = hi ? tw   : qt.y;
        Bt1.x = hi ? qt.z : tx;    Bt1.y = hi ? qt.w : ty;

        // 4 WMMAs per 32 pixels; A (gt) frag shared by both accumulators
        accS = __builtin_amdgcn_wmma_f32_16x16x4_f32(false, A0, false, Bs0,
                                                     (short)0, accS, false, false);
        accT = __builtin_amdgcn_wmma_f32_16x16x4_f32(false, A0, false, Bt0,
                                                     (short)0, accT, false, false);
        accS = __builtin_amdgcn_wmma_f32_16x16x4_f32(false, A1, false, Bs1,
                                                     (short)0, accS, false, false);
        accT = __builtin_amdgcn_wmma_f32_16x16x4_f32(false, A1, false, Bt1,
                                                     (short)0, accT, false, false);
    }

    // Extract diagonal 4x4 blocks of the accumulators.
    // C/D layout: lo lane holds C[r][N=lane] in acc[r]; hi lane holds C[r+8][N=lane-16].
    const int  N  = lane & 15;
    const int  n2 = N >> 2;        // column block s
    const int  cc = N & 3;         // logit channel
    const bool contrib = hi ? (n2 >= 2) : (n2 < 2);
    if (contrib) {
        const bool upper = (n2 & 1) != 0;   // which half of acc holds M=4*n2+j
#pragma unroll
        for (int j = 0; j < 4; ++j) {       // j == class index i
            const float vs = upper ? accS[j + 4] : accS[j];
            const float vt = upper ? accT[j + 4] : accT[j];
            atomicAdd(&sred[j * 4 + cc], vs);
            atomicAdd(&sred[16 + j * 4 + cc], vt);
        }
    }
    atomicAdd(&sred[32 + ch], cnt);
    __syncthreads();

    if (tid < 36) atomicAdd(&ws[tid], sred[tid]);
}

__global__ void kd_finalize(const float* __restrict__ ws, float* __restrict__ out) {
    if (threadIdx.x != 0 || blockIdx.x != 0) return;
    float a[16], bb[16], p[16], q[16];
#pragma unroll
    for (int i = 0; i < 4; ++i) {
        const float cinv = 0.5f / (ws[32 + i] + 1e-6f);  // /count, /temperature(2)
#pragma unroll
        for (int c = 0; c < 4; ++c) {
            a[i * 4 + c]  = ws[i * 4 + c] * cinv;
            bb[i * 4 + c] = ws[16 + i * 4 + c] * cinv;
        }
    }
#pragma unroll
    for (int i = 0; i < 4; ++i) {   // row softmaxes
        float ma = a[i * 4], mb = bb[i * 4];
        for (int c = 1; c < 4; ++c) {
            ma = fmaxf(ma, a[i * 4 + c]);
            mb = fmaxf(mb, bb[i * 4 + c]);
        }
        float sa = 0.0f, sb = 0.0f;
        for (int c = 0; c < 4; ++c) {
            p[i * 4 + c] = expf(a[i * 4 + c] - ma);  sa += p[i * 4 + c];
            q[i * 4 + c] = expf(bb[i * 4 + c] - mb); sb += q[i * 4 + c];
        }
        for (int c = 0; c < 4; ++c) { p[i * 4 + c] /= sa; q[i * 4 + c] /= sb; }
    }
    float sym = 0.0f;
    for (int k = 0; k < 16; ++k)
        sym += 0.5f * (p[k] * logf(p[k] / q[k]) + q[k] * logf(q[k] / p[k]));
    out[0] = sym * 0.25f;   // / num_classes
}

extern "C" void kernel_launch(void* const* d_in, const int* in_sizes, int n_in,
                              void* d_out, int out_size, void* d_ws, size_t ws_size,
                              hipStream_t stream) {
    const float* src = (const float*)d_in[0];
    const float* tgt = (const float*)d_in[1];
    const float* gt  = (const float*)d_in[2];
    // d_in[3] = num_classes (fixed at 4 here)
    float* ws  = (float*)d_ws;
    float* out = (float*)d_out;

    kd_zero_ws<<<1, 64, 0, stream>>>(ws);
    kd_sums_kernel<<<KD_BLOCKS, KD_THREADS, 0, stream>>>(src, tgt, gt, ws);
    kd_finalize<<<1, 32, 0, stream>>>(ws, out);
}